// DistoLayer_80384607912002
// MI455X (gfx1250) — compile-verified
//
#include <hip/hip_runtime.h>
#include <hip/hip_bf16.h>

// N=100000, M=10000, K=32, D=64, H=8, C=64
// Restructured: pair[m,k,h] = x_src[m,k] . (x_dst[m] @ Wcomb_h), Wcomb = f(Wq,Wk)

typedef float v2f __attribute__((ext_vector_type(2)));
typedef float v8f __attribute__((ext_vector_type(8)));
typedef int   v4i_ __attribute__((ext_vector_type(4)));

#if defined(__gfx1250__) && __has_builtin(__builtin_amdgcn_global_load_async_to_lds_b128) && __has_builtin(__builtin_amdgcn_s_wait_asynccnt)
#define USE_ASYNC_LDS 1
#else
#define USE_ASYNC_LDS 0
#endif

__device__ __forceinline__ v8f wmma_f32_4(v2f a, v2f b, v8f c) {
  // V_WMMA_F32_16X16X4_F32 : D[16x16] = A[16x4] * B[4x16] + C
  return __builtin_amdgcn_wmma_f32_16x16x4_f32(
      /*neg_a=*/false, a, /*neg_b=*/false, b,
      /*c_mod=*/(short)0, c, /*reuse_a=*/false, /*reuse_b=*/false);
}

// ---------------------------------------------------------------------------
// Kernel 1: Wcomb[a][h*64+b] = sum_d Wq[a,d] * Wk[b, h*64+d]   (64 x 512)
// ---------------------------------------------------------------------------
__global__ __launch_bounds__(256) void k_wcomb(const float* __restrict__ Wq,
                                               const float* __restrict__ Wk,
                                               float* __restrict__ Wcomb) {
  int j = blockIdx.x * 256 + threadIdx.x;   // 0 .. 32767
  int a  = j >> 9;
  int hb = j & 511;
  int h  = hb >> 6;
  int b  = hb & 63;
  const float* wq = Wq + a * 64;
  const float* wk = Wk + (size_t)b * 512 + h * 64;
  float s = 0.f;
#pragma unroll
  for (int d = 0; d < 64; ++d) s += wq[d] * wk[d];
  Wcomb[(size_t)a * 512 + hb] = s;
}

// ---------------------------------------------------------------------------
// Kernel 2: G[m][j] = sum_a x[dst_idx[m], a] * Wcomb[a, j]   ([M,64]@[64,512])
// One 16x16 output tile per wave, 16 WMMA k-steps. 8 waves/block.
// ---------------------------------------------------------------------------
__global__ __launch_bounds__(256) void k_gmat(const float* __restrict__ x,
                                              const int* __restrict__ dst_idx,
                                              const float* __restrict__ Wcomb,
                                              float* __restrict__ G, int M) {
  __shared__ float Xd[16][68];   // padded: bank-conflict-free column reads
  __shared__ int   rows[16];
  const int t  = threadIdx.x;
  const int mt = blockIdx.x;     // m-tile (16 rows)
  const int nb = blockIdx.y;     // 0..3, 128 cols each

  if (t < 16) {
    int m = mt * 16 + t;
    rows[t] = (m < M) ? dst_idx[m] : 0;
  }
  __syncthreads();
#if USE_ASYNC_LDS
  {
    // 16 rows x 64 floats = 256 B128 transfers; 1 per thread
    int r = t >> 4, c = (t & 15) * 4;
    __builtin_amdgcn_global_load_async_to_lds_b128(
        (v4i_*)(x + (size_t)rows[r] * 64 + c), (v4i_*)&Xd[r][c], 0, 0);
    __builtin_amdgcn_s_wait_asynccnt(0);
  }
#else
  for (int i = t; i < 16 * 64; i += 256) {
    int r = i >> 6, c = i & 63;
    Xd[r][c] = x[(size_t)rows[r] * 64 + c];
  }
#endif
  __syncthreads();

  const int wave = t >> 5, lane = t & 31;
  const int half = lane >> 4, lm = lane & 15;
  const int j0 = nb * 128 + wave * 16;

  v8f acc = {};
#pragma unroll
  for (int s = 0; s < 16; ++s) {
    int ka = 4 * s + 2 * half;          // K rows held by this lane half
    v2f a, b;
    a.x = Xd[lm][ka];
    a.y = Xd[lm][ka + 1];
    b.x = Wcomb[(size_t)ka * 512 + j0 + lm];
    b.y = Wcomb[(size_t)(ka + 1) * 512 + j0 + lm];
    acc = wmma_f32_4(a, b, acc);
  }
  if (mt * 16 + 16 <= M) {               // uniform full-tile fast path
#pragma unroll
    for (int r = 0; r < 8; ++r) {
      G[(size_t)(mt * 16 + r + 8 * half) * 512 + j0 + lm] = acc[r];
    }
  } else {
#pragma unroll
    for (int r = 0; r < 8; ++r) {
      int m = mt * 16 + r + 8 * half;
      if (m < M) G[(size_t)m * 512 + j0 + lm] = acc[r];
    }
  }
}

// ---------------------------------------------------------------------------
// Kernel 3: per m (one block, 2 waves):
//   Xs[32,64] = gather(x, src_idx[m,:])
//   pair[32,8]  = (Xs @ Gm^T) / 8            (WMMA, 2 tiles x 16 steps)
//   hid[32,8]   = relu(pair @ Wp1)           (scalar, tiny)
//   out[32,64]  = hid @ Wp2                  (WMMA, 8 tiles x 2 steps)
// ---------------------------------------------------------------------------
__global__ __launch_bounds__(64) void k_attn(const float* __restrict__ x,
                                             const int* __restrict__ src_idx,
                                             const float* __restrict__ G,
                                             const float* __restrict__ Wp1,
                                             const float* __restrict__ Wp2,
                                             float* __restrict__ out, int M) {
  __shared__ float GmL[16][68];   // rows 8..15 zero (WMMA N-dim padding)
  __shared__ float Xs[32][68];
  __shared__ float pairL[32][9];
  __shared__ float hidL[32][9];
  __shared__ float Wp1L[8][8];
  __shared__ float Wp2L[8][68];
  __shared__ int   srows[32];

  const int m = blockIdx.x;
  const int t = threadIdx.x;      // 0..63

  Wp1L[t >> 3][t & 7] = Wp1[t];                       // 64 elems, 64 threads
  for (int i = t; i < 8 * 64; i += 64) Wp2L[i >> 6][i & 63] = Wp2[i];
  if (t < 32) srows[t] = src_idx[(size_t)m * 32 + t];
  // zero the padded upper half of GmL (rows 8..15)
  for (int i = t; i < 8 * 64; i += 64) GmL[8 + (i >> 6)][i & 63] = 0.f;
  __syncthreads();

#if USE_ASYNC_LDS
  {
    // Gm rows 0..7: 512 floats = 128 B128 ops; 2 per thread
#pragma unroll
    for (int j = 0; j < 2; ++j) {
      int f = (t + j * 64) * 4;
      int h = f >> 6, b = f & 63;
      __builtin_amdgcn_global_load_async_to_lds_b128(
          (v4i_*)(G + (size_t)m * 512 + f), (v4i_*)&GmL[h][b], 0, 0);
    }
    // Xs: 32 rows x 64 floats = 512 B128 ops; 8 per thread
#pragma unroll
    for (int j = 0; j < 8; ++j) {
      int f = (t + j * 64) * 4;
      int k = f >> 6, c = f & 63;
      __builtin_amdgcn_global_load_async_to_lds_b128(
          (v4i_*)(x + (size_t)srows[k] * 64 + c), (v4i_*)&Xs[k][c], 0, 0);
    }
    __builtin_amdgcn_s_wait_asynccnt(0);
  }
#else
  for (int i = t; i < 8 * 64; i += 64) {
    int h = i >> 6, b = i & 63;
    GmL[h][b] = G[(size_t)m * 512 + i];
  }
  for (int i = t; i < 32 * 64; i += 64) {
    int k = i >> 6, c = i & 63;
    Xs[k][c] = x[(size_t)srows[k] * 64 + c];
  }
#endif
  __syncthreads();

  const int wave = t >> 5, lane = t & 31;
  const int half = lane >> 4, lm = lane & 15;

  // ---- pair = Xs @ Gm^T (k-tile per wave; N = h, only 8 valid cols) ----
  {
    v8f acc = {};
#pragma unroll
    for (int s = 0; s < 16; ++s) {
      int kb = 4 * s + 2 * half;       // b-dim rows for this lane half
      v2f a, b;
      a.x = Xs[wave * 16 + lm][kb];
      a.y = Xs[wave * 16 + lm][kb + 1];
      b.x = GmL[lm][kb];               // B[kb][h=lm] = Gm[h*64+kb]
      b.y = GmL[lm][kb + 1];
      acc = wmma_f32_4(a, b, acc);
    }
#pragma unroll
    for (int r = 0; r < 8; ++r) {
      if (lm < 8) pairL[wave * 16 + r + 8 * half][lm] = acc[r] * 0.125f;
    }
  }
  __syncthreads();

  // ---- hidden = relu(pair @ Wp1), 256 tiny outputs ----
  for (int o = t; o < 256; o += 64) {
    int k = o >> 3, hh = o & 7;
    float s = 0.f;
#pragma unroll
    for (int h2 = 0; h2 < 8; ++h2) s += pairL[k][h2] * Wp1L[h2][hh];
    hidL[k][hh] = s > 0.f ? s : 0.f;
  }
  __syncthreads();

  // ---- logits = hidden @ Wp2 : [32,8]@[8,64]; wave w -> k-tile w, 4 c-tiles
  for (int ct = 0; ct < 4; ++ct) {
    v8f acc = {};
#pragma unroll
    for (int s = 0; s < 2; ++s) {
      int hb2 = 4 * s + 2 * half;      // hidden-dim rows for this lane half
      v2f a, b;
      a.x = hidL[wave * 16 + lm][hb2];
      a.y = hidL[wave * 16 + lm][hb2 + 1];
      b.x = Wp2L[hb2][ct * 16 + lm];
      b.y = Wp2L[hb2 + 1][ct * 16 + lm];
      acc = wmma_f32_4(a, b, acc);
    }
    size_t base = ((size_t)m * 32 + wave * 16 + 8 * half) * 64 + ct * 16 + lm;
#pragma unroll
    for (int r = 0; r < 8; ++r) out[base + (size_t)r * 64] = acc[r];
  }
}

// ---------------------------------------------------------------------------
extern "C" void kernel_launch(void* const* d_in, const int* in_sizes, int n_in,
                              void* d_out, int out_size, void* d_ws, size_t ws_size,
                              hipStream_t stream) {
  const float* x       = (const float*)d_in[0];
  const int*   src_idx = (const int*)d_in[1];
  const int*   dst_idx = (const int*)d_in[2];
  const float* Wq      = (const float*)d_in[3];
  const float* Wk      = (const float*)d_in[4];
  const float* Wp1     = (const float*)d_in[5];
  const float* Wp2     = (const float*)d_in[6];
  float* out = (float*)d_out;
  const int M = in_sizes[2];              // 10000

  float* Wcomb = (float*)d_ws;            // 64*512 floats = 128 KB
  float* G     = Wcomb + 64 * 512;        // M*512 floats ~ 20.5 MB

  k_wcomb<<<128, 256, 0, stream>>>(Wq, Wk, Wcomb);
  dim3 g2((M + 15) / 16, 4);
  k_gmat<<<g2, 256, 0, stream>>>(x, dst_idx, Wcomb, G, M);
  k_attn<<<M, 64, 0, stream>>>(x, src_idx, G, Wp1, Wp2, out, M);
}